// QDQN_9740985827976
// MI455X (gfx1250) — compile-verified
//
#include <hip/hip_runtime.h>
#include <cstddef>
#include <cstdint>

// ---------------------------------------------------------------------------
// 20-qubit statevector QNN forward for MI455X (gfx1250, wave32).
//
// Design:
//  * State: 16 batches x 2^20 complex64 (float2, SoA-in-pair) = 128 MB in d_ws.
//    Fits MI455X's 192 MB L2 -> all gate passes run at L2 bandwidth.
//  * CNOT rings are linear maps L over GF(2)^20 on index bits; they are never
//    applied physically. Layer-m gates are conjugated through P^m: a 4-qubit
//    group gate couples cosets x ^ span(cols of L^-m), with gate row labels
//    a_j = parity(x & row_(4g+j)(L^m)).  All masks computed host-side.
//  * Each layer's 20 U3 gates fuse into five 16x16 complex unitaries (kron of
//    four 2x2s). Applying one = complex GEMM U(16x16) x S(16 x 16 cosets)
//    = 16 x V_WMMA_F32_16X16X4_F32 per wave iteration, fully in-place.
//  * Layer 0 (RX encode + U3) on |0..0> yields a product state -> built by a
//    single streaming pass from two 1024-entry partial-product tables.
//  * Measurement <Z0Z1Z2> = sum |amp|^2 * (-1)^popc(x & maskZ),
//    maskZ = r0^r1^r2 of L^4.  Deterministic 2-stage tree reduction.
// ---------------------------------------------------------------------------

typedef float v2f __attribute__((ext_vector_type(2)));
typedef float v8f __attribute__((ext_vector_type(8)));

#define NQ 20
#define NSTATE (1u << NQ)
#define BATCH 16

struct PassArgs {
    unsigned rm[4];      // rows 4g+j of L^m  (gate row-label parities)
    unsigned v[4];       // columns 4g+j of (L^m)^-1 (coset XOR generators)
    unsigned compMask;   // 16 bit-positions forming a complement basis
};

// ----------------------------- device helpers ------------------------------

__device__ __forceinline__ float2 cmul(float2 a, float2 b) {
    float2 r; r.x = a.x * b.x - a.y * b.y; r.y = a.x * b.y + a.y * b.x; return r;
}
__device__ __forceinline__ float2 cadd(float2 a, float2 b) {
    float2 r; r.x = a.x + b.x; r.y = a.y + b.y; return r;
}
__device__ __forceinline__ unsigned combo4(unsigned a, unsigned v0, unsigned v1,
                                           unsigned v2, unsigned v3) {
    unsigned r = 0;
    if (a & 1u) r ^= v0;
    if (a & 2u) r ^= v1;
    if (a & 4u) r ^= v2;
    if (a & 8u) r ^= v3;
    return r;
}
__device__ __forceinline__ float2 u3_elem(float th, float ph, float de, int i, int j) {
    float c = __cosf(th * 0.5f), s = __sinf(th * 0.5f);
    float2 r;
    if (i == 0 && j == 0) { r.x = c; r.y = 0.f; }
    else if (i == 0)      { r.x = -__cosf(de) * s; r.y = -__sinf(de) * s; }
    else if (j == 0)      { r.x =  __cosf(ph) * s; r.y =  __sinf(ph) * s; }
    else                  { r.x = __cosf(ph + de) * c; r.y = __sinf(ph + de) * c; }
    return r;
}

// ------------------------------ prep kernel --------------------------------
// Computes (a) layer-0 per-batch 2-vectors v[b][w] = col0 of U3*RX,
//          (b) fifteen 16x16 complex group unitaries for layers 1..3,
//          (c) per-batch product tables Tlo/Thi (1024 entries each).
__global__ __launch_bounds__(256)
void prep_kernel(const float* __restrict__ inp,
                 const float* __restrict__ thw,
                 const float* __restrict__ phw,
                 const float* __restrict__ dew,
                 float2* __restrict__ vcol,   // [16][20][2]
                 float2* __restrict__ Uall,   // [15][256]
                 float2* __restrict__ tab)    // [16][2048] : lo then hi
{
    const int tid = threadIdx.x;

    // phase 1: layer-0 columns (U3 * RX applied to |0>)
    for (int t = tid; t < BATCH * NQ; t += 256) {
        int b = t / NQ, w = t % NQ;
        int wi = w * 12 + 5;  // layer 0, block p=0, op index 5
        float th = thw[wi], ph = phw[wi], de = dew[wi];
        float2 u00 = u3_elem(th, ph, de, 0, 0);
        float2 u01 = u3_elem(th, ph, de, 0, 1);
        float2 u10 = u3_elem(th, ph, de, 1, 0);
        float2 u11 = u3_elem(th, ph, de, 1, 1);
        float x = inp[b * NQ + w];
        float cr = __cosf(x * 0.5f), sr = __sinf(x * 0.5f);
        float2 r0; r0.x = cr;  r0.y = 0.f;   // RX col0 row0 = c
        float2 r1; r1.x = 0.f; r1.y = -sr;   // RX col0 row1 = -i s
        vcol[b * 40 + w * 2 + 0] = cadd(cmul(u00, r0), cmul(u01, r1));
        vcol[b * 40 + w * 2 + 1] = cadd(cmul(u10, r0), cmul(u11, r1));
    }

    // phase 2: 16x16 group unitaries, layers m=1..3, groups g=0..4
    for (int idx = tid; idx < 15 * 256; idx += 256) {
        int p = idx >> 8;            // pass index
        int m = p / 5 + 1, g = p % 5;
        int r = (idx >> 4) & 15, c = idx & 15;
        float2 prod; prod.x = 1.f; prod.y = 0.f;
        for (int j = 0; j < 4; ++j) {
            int w  = g * 4 + j;
            int wi = m * 240 + w * 12 + 5;
            float2 e = u3_elem(thw[wi], phw[wi], dew[wi], (r >> j) & 1, (c >> j) & 1);
            prod = cmul(prod, e);
        }
        Uall[p * 256 + r * 16 + c] = prod;
    }

    __syncthreads();

    // phase 3: product-state tables: Tlo over qubits 0..9, Thi over 10..19
    for (int idx = tid; idx < BATCH * 2048; idx += 256) {
        int b = idx >> 11;
        int t = idx & 2047;
        float2 prod; prod.x = 1.f; prod.y = 0.f;
        if (t < 1024) {
            for (int w = 0; w < 10; ++w)
                prod = cmul(prod, vcol[b * 40 + w * 2 + ((t >> w) & 1)]);
        } else {
            unsigned tt = (unsigned)(t - 1024);
            for (int w = 10; w < 20; ++w)
                prod = cmul(prod, vcol[b * 40 + w * 2 + ((tt >> (w - 10)) & 1)]);
        }
        tab[idx] = prod;
    }
}

// ----------------------------- init state ----------------------------------
__global__ __launch_bounds__(256)
void init_state_kernel(float2* __restrict__ state, const float2* __restrict__ tab)
{
    size_t gid = (size_t)blockIdx.x * blockDim.x + threadIdx.x;  // 16M threads
    unsigned b = (unsigned)(gid >> NQ);
    unsigned x = (unsigned)gid & (NSTATE - 1u);
    float2 lo = tab[b * 2048 + (x & 1023u)];
    float2 hi = tab[b * 2048 + 1024u + (x >> 10)];
    state[gid] = cmul(hi, lo);
}

// --------------------------- WMMA gate pass --------------------------------
// One pass applies one 16x16 complex unitary, in-place, to all 16 batches.
// A wave handles 16 cosets per iteration:
//   B(16x16) = gathered state values (row = gate input label k, col = coset n)
//   D = U * B  via 16x V_WMMA_F32_16X16X4_F32 (4 real GEMMs, K split 4x4).
__global__ __launch_bounds__(256)
void gate_pass_kernel(float2* __restrict__ state,
                      const float2* __restrict__ U,
                      PassArgs pa)
{
    __builtin_prefetch(U, 0, 1);  // 2KB gate matrix -> global_prefetch_b8

    const int lane = (int)(threadIdx.x & 31u);   // wave32
    const int n    = lane & 15;                  // coset / column index
    const int half = lane >> 4;                  // 0: K/M low, 1: K/M high

    // A-matrix fragments (16x4 f32 layout: x->K=4kb+2h, y->K=4kb+2h+1; M=lane%16)
    v2f ar[4], ai[4], ani[4];
#pragma unroll
    for (int kb = 0; kb < 4; ++kb) {
        int k0 = kb * 4 + half * 2;
        float2 u0 = U[n * 16 + k0];
        float2 u1 = U[n * 16 + k0 + 1];
        v2f t;
        t.x = u0.x;  t.y = u1.x;  ar[kb]  = t;
        t.x = u0.y;  t.y = u1.y;  ai[kb]  = t;
        t.x = -u0.y; t.y = -u1.y; ani[kb] = t;   // f32 WMMA has no A-neg: bake it in
    }

    const unsigned v0 = pa.v[0], v1 = pa.v[1], v2 = pa.v[2], v3 = pa.v[3];

    const int wave   = (int)(blockIdx.x * (blockDim.x >> 5) + (threadIdx.x >> 5));
    const int nWaves = (int)(gridDim.x * (blockDim.x >> 5));
    const int totalChunks = BATCH * (int)(NSTATE / 16 / 16);  // 16 * 4096

    for (int chunk = wave; chunk < totalChunks; chunk += nWaves) {
        const int b = chunk >> 12;
        const unsigned cidx = ((unsigned)(chunk & 4095) << 4) | (unsigned)n;

        // spread the 16 coset-counter bits into the complement bit positions
        unsigned rep = 0;
        {
            unsigned mask = pa.compMask, t = cidx;
            while (mask) {
                unsigned bit = mask & (0u - mask);
                rep |= (t & 1u) ? bit : 0u;
                t >>= 1u;
                mask &= mask - 1u;
            }
        }
        // gate-row label of rep, then move to the coset member with label 0
        const unsigned a0 =
              (unsigned)(__popc(rep & pa.rm[0]) & 1)
            | ((unsigned)(__popc(rep & pa.rm[1]) & 1) << 1)
            | ((unsigned)(__popc(rep & pa.rm[2]) & 1) << 2)
            | ((unsigned)(__popc(rep & pa.rm[3]) & 1) << 3);
        const unsigned base = rep ^ combo4(a0, v0, v1, v2, v3);

        float2* __restrict__ S = state + ((size_t)b << NQ);

        // gather B fragments (8 x 8-byte loads per lane)
        v2f br[4], bi[4];
#pragma unroll
        for (int kb = 0; kb < 4; ++kb) {
            unsigned k0 = (unsigned)(kb * 4 + half * 2);
            float2 s0 = S[base ^ combo4(k0,      v0, v1, v2, v3)];
            float2 s1 = S[base ^ combo4(k0 + 1u, v0, v1, v2, v3)];
            v2f t;
            t.x = s0.x; t.y = s1.x; br[kb] = t;
            t.x = s0.y; t.y = s1.y; bi[kb] = t;
        }

        // complex GEMM: Cr = Ur*Br - Ui*Bi ; Ci = Ur*Bi + Ui*Br
        v8f cr = {0.f, 0.f, 0.f, 0.f, 0.f, 0.f, 0.f, 0.f};
        v8f ci = {0.f, 0.f, 0.f, 0.f, 0.f, 0.f, 0.f, 0.f};
#pragma unroll
        for (int kb = 0; kb < 4; ++kb) {
            cr = __builtin_amdgcn_wmma_f32_16x16x4_f32(false, ar[kb],  false, br[kb], (short)0, cr, false, false);
            cr = __builtin_amdgcn_wmma_f32_16x16x4_f32(false, ani[kb], false, bi[kb], (short)0, cr, false, false);
            ci = __builtin_amdgcn_wmma_f32_16x16x4_f32(false, ar[kb],  false, bi[kb], (short)0, ci, false, false);
            ci = __builtin_amdgcn_wmma_f32_16x16x4_f32(false, ai[kb],  false, br[kb], (short)0, ci, false, false);
        }

        // scatter D: VGPR r holds M=r (lanes 0-15) / M=r+8 (lanes 16-31), N=n
#pragma unroll
        for (int rr = 0; rr < 8; ++rr) {
            unsigned M = (unsigned)(rr + half * 8);
            float2 o; o.x = cr[rr]; o.y = ci[rr];
            S[base ^ combo4(M, v0, v1, v2, v3)] = o;
        }
    }
}

// ----------------------------- measurement ---------------------------------
__global__ __launch_bounds__(256)
void measure_partial_kernel(const float2* __restrict__ state,
                            float* __restrict__ partial, unsigned maskZ)
{
    const int blk = (int)blockIdx.x;       // 16*64 blocks
    const int b   = blk >> 6;
    const int sub = blk & 63;
    float acc = 0.f;
    for (int i = (int)threadIdx.x; i < 16384; i += 256) {
        unsigned x = ((unsigned)sub << 14) + (unsigned)i;
        float2 a = state[((size_t)b << NQ) + x];
        float p = a.x * a.x + a.y * a.y;
        acc += (__popc(x & maskZ) & 1) ? -p : p;
    }
    __shared__ float red[256];
    red[threadIdx.x] = acc;
    __syncthreads();
    for (int s = 128; s > 0; s >>= 1) {
        if ((int)threadIdx.x < s) red[threadIdx.x] += red[threadIdx.x + s];
        __syncthreads();
    }
    if (threadIdx.x == 0) partial[blk] = red[0];
}

__global__ __launch_bounds__(64)
void measure_final_kernel(const float* __restrict__ partial, float* __restrict__ out)
{
    __shared__ float red[64];
    red[threadIdx.x] = partial[blockIdx.x * 64 + threadIdx.x];
    __syncthreads();
    for (int s = 32; s > 0; s >>= 1) {
        if ((int)threadIdx.x < s) red[threadIdx.x] += red[threadIdx.x + s];
        __syncthreads();
    }
    if (threadIdx.x == 0) out[blockIdx.x] = red[0];
}

// --------------------------- host GF(2) algebra ----------------------------

struct BitMat { unsigned r[NQ]; };

static BitMat bm_identity() {
    BitMat I;
    for (int i = 0; i < NQ; ++i) I.r[i] = 1u << i;
    return I;
}
static BitMat bm_mul(const BitMat& A, const BitMat& B) {
    BitMat C;
    for (int i = 0; i < NQ; ++i) {
        unsigned m = 0;
        for (int j = 0; j < NQ; ++j)
            if ((A.r[i] >> j) & 1u) m ^= B.r[j];
        C.r[i] = m;
    }
    return C;
}
static BitMat bm_inv(BitMat M) {
    BitMat I = bm_identity();
    for (int col = 0; col < NQ; ++col) {
        int p = col;
        while (p < NQ && !((M.r[p] >> col) & 1u)) ++p;
        if (p >= NQ) continue;  // (ring map is invertible; not reached)
        unsigned tm = M.r[col]; M.r[col] = M.r[p]; M.r[p] = tm;
        unsigned ti = I.r[col]; I.r[col] = I.r[p]; I.r[p] = ti;
        for (int row = 0; row < NQ; ++row)
            if (row != col && ((M.r[row] >> col) & 1u)) {
                M.r[row] ^= M.r[col];
                I.r[row] ^= I.r[col];
            }
    }
    return I;
}
static unsigned bm_col(const BitMat& A, int c) {
    unsigned m = 0;
    for (int row = 0; row < NQ; ++row)
        if ((A.r[row] >> c) & 1u) m |= 1u << row;
    return m;
}
// find 16 unit-vector positions completing {v0..v3} to a basis of GF(2)^20
static unsigned find_complement(const unsigned v[4]) {
    unsigned piv[NQ] = {0};
    auto add = [&](unsigned x) -> bool {
        for (int b = NQ - 1; b >= 0; --b) {
            if ((x >> b) & 1u) {
                if (piv[b]) x ^= piv[b];
                else { piv[b] = x; return true; }
            }
        }
        return false;
    };
    for (int j = 0; j < 4; ++j) add(v[j]);
    unsigned comp = 0; int cnt = 0;
    for (int b = 0; b < NQ && cnt < 16; ++b)
        if (add(1u << b)) { comp |= 1u << b; ++cnt; }
    return comp;
}

// ------------------------------- launcher ----------------------------------

extern "C" void kernel_launch(void* const* d_in, const int* in_sizes, int n_in,
                              void* d_out, int out_size, void* d_ws, size_t ws_size,
                              hipStream_t stream)
{
    (void)in_sizes; (void)n_in; (void)out_size;

    const float* inp = (const float*)d_in[0];
    const float* thw = (const float*)d_in[1];
    const float* phw = (const float*)d_in[2];
    const float* dew = (const float*)d_in[3];
    float* out = (float*)d_out;

    // workspace layout
    const size_t OFF_STATE = 0;
    const size_t SZ_STATE  = (size_t)BATCH * NSTATE * sizeof(float2);   // 128 MB
    const size_t OFF_U     = OFF_STATE + SZ_STATE;
    const size_t SZ_U      = 15 * 256 * sizeof(float2);
    const size_t OFF_VCOL  = OFF_U + SZ_U;
    const size_t SZ_VCOL   = (size_t)BATCH * 40 * sizeof(float2);
    const size_t OFF_TAB   = OFF_VCOL + SZ_VCOL;
    const size_t SZ_TAB    = (size_t)BATCH * 2048 * sizeof(float2);
    const size_t OFF_PART  = OFF_TAB + SZ_TAB;
    const size_t SZ_PART   = (size_t)BATCH * 64 * sizeof(float);
    if (ws_size < OFF_PART + SZ_PART) return;   // cannot run without state buffer

    char* ws = (char*)d_ws;
    float2* state   = (float2*)(ws + OFF_STATE);
    float2* Uall    = (float2*)(ws + OFF_U);
    float2* vcol    = (float2*)(ws + OFF_VCOL);
    float2* tab     = (float2*)(ws + OFF_TAB);
    float*  partial = (float*)(ws + OFF_PART);

    // CNOT-ring linear map: b_t' = XOR(b_0..b_t) for t>=1; b_0' = XOR(b_1..b_19)
    BitMat L;
    L.r[0] = 0xFFFFEu;
    for (int t = 1; t < NQ; ++t) L.r[t] = (1u << (t + 1)) - 1u;
    BitMat Lp[5];
    Lp[1] = L;
    Lp[2] = bm_mul(L, Lp[1]);
    Lp[3] = bm_mul(L, Lp[2]);
    Lp[4] = bm_mul(L, Lp[3]);

    // 1) small prep (gates + product tables)
    prep_kernel<<<1, 256, 0, stream>>>(inp, thw, phw, dew, vcol, Uall, tab);

    // 2) build product state after layer 0 (single streaming write of 128 MB)
    init_state_kernel<<<(BATCH * NSTATE) / 256, 256, 0, stream>>>(state, tab);

    // 3) fifteen in-place WMMA gate passes (layers 1..3 x 5 qubit groups);
    //    CNOT rings are folded into the coset addressing (zero extra passes)
    for (int m = 1; m <= 3; ++m) {
        BitMat inv = bm_inv(Lp[m]);
        for (int g = 0; g < 5; ++g) {
            PassArgs pa;
            for (int j = 0; j < 4; ++j) {
                pa.rm[j] = Lp[m].r[4 * g + j];
                pa.v[j]  = bm_col(inv, 4 * g + j);
            }
            pa.compMask = find_complement(pa.v);
            const int pass = (m - 1) * 5 + g;
            gate_pass_kernel<<<1024, 256, 0, stream>>>(state, Uall + pass * 256, pa);
        }
    }

    // 4) <Z0 Z1 Z2> with all four rings folded into one parity mask
    unsigned maskZ = Lp[4].r[0] ^ Lp[4].r[1] ^ Lp[4].r[2];
    measure_partial_kernel<<<BATCH * 64, 256, 0, stream>>>(state, partial, maskZ);
    measure_final_kernel<<<BATCH, 64, 0, stream>>>(partial, out);
}